// Group_32882269618428
// MI455X (gfx1250) — compile-verified
//
#include <hip/hip_runtime.h>
#include <hip/hip_bf16.h>
#include <math.h>

// ---------------------------------------------------------------------------
// Problem constants (from reference): B=16, N=8192, G=512 centers, M=32 knn.
// Output layout (flattened tuple, return order):
//   neighborhood [B*G*M*3] | center [B*G*3] | neighborhood_org [B*G*M*3]
// ---------------------------------------------------------------------------
#define BATCH      16
#define NPTS       8192
#define NGROUP     512
#define KNN        32
#define CHUNK      256                 // points scored per LDS chunk in KNN
#define NCHUNK     (NPTS / CHUNK)      // 32
#define CAP        CHUNK               // per-wave candidate buffer capacity
#define NEIGH_ELEMS  (BATCH * NGROUP * KNN * 3)   // 786432
#define CENTER_ELEMS (BATCH * NGROUP * 3)         // 24576

typedef __attribute__((ext_vector_type(2))) float v2f;
typedef __attribute__((ext_vector_type(8))) float v8f;

// ---------------------------------------------------------------------------
// Kernel 1: farthest point sampling. One workgroup per batch, 1024 threads.
// Each thread owns 8 points (coords + running min-dist live in VGPRs).
// Per iteration: block-wide argmax (tie -> lowest index, matches jnp.argmax),
// winner broadcasts its point through LDS and writes the center to global.
// ---------------------------------------------------------------------------
__global__ __launch_bounds__(1024) void fps_kernel(const float* __restrict__ xyz,
                                                   float* __restrict__ centerOut) {
  __shared__ float sWaveV[32];
  __shared__ int   sWaveI[32];
  __shared__ int   sPickIdx;
  __shared__ float sPickX, sPickY, sPickZ;

  const int b = blockIdx.x;
  const int t = threadIdx.x;
  const float* xb = xyz + (size_t)b * NPTS * 3;

  float px[8], py[8], pz[8], md[8];
#pragma unroll
  for (int j = 0; j < 8; ++j) {
    const int p = t + j * 1024;
    px[j] = xb[p * 3 + 0];
    py[j] = xb[p * 3 + 1];
    pz[j] = xb[p * 3 + 2];
    md[j] = INFINITY;
  }

  for (int k = 0; k < NGROUP; ++k) {
    // thread-local argmax over the 8 owned points (ascending index => first max)
    float v = -INFINITY;
    int   i = 0x7fffffff;
#pragma unroll
    for (int j = 0; j < 8; ++j) {
      if (md[j] > v) { v = md[j]; i = t + j * 1024; }
    }
    // wave32 butterfly argmax, tie -> lower index
    for (int off = 16; off > 0; off >>= 1) {
      const float ov = __shfl_xor(v, off, 32);
      const int   oi = __shfl_xor(i, off, 32);
      if (ov > v || (ov == v && oi < i)) { v = ov; i = oi; }
    }
    if ((t & 31) == 0) { sWaveV[t >> 5] = v; sWaveI[t >> 5] = i; }
    __syncthreads();
    if (t < 32) {
      float v2 = sWaveV[t];
      int   i2 = sWaveI[t];
      for (int off = 16; off > 0; off >>= 1) {
        const float ov = __shfl_xor(v2, off, 32);
        const int   oi = __shfl_xor(i2, off, 32);
        if (ov > v2 || (ov == v2 && oi < i2)) { v2 = ov; i2 = oi; }
      }
      if (t == 0) sPickIdx = i2;
    }
    __syncthreads();
    const int win = sPickIdx;
    if ((win & 1023) == t) {           // owning thread broadcasts + writes center
      const int jj = win >> 10;
      float wx = 0.f, wy = 0.f, wz = 0.f;
#pragma unroll
      for (int j = 0; j < 8; ++j)
        if (j == jj) { wx = px[j]; wy = py[j]; wz = pz[j]; }
      sPickX = wx; sPickY = wy; sPickZ = wz;
      float* co = centerOut + ((size_t)(b * NGROUP + k)) * 3;
      co[0] = wx; co[1] = wy; co[2] = wz;
    }
    __syncthreads();
    const float cx = sPickX, cy = sPickY, cz = sPickZ;
#pragma unroll
    for (int j = 0; j < 8; ++j) {
      const float dx = px[j] - cx, dy = py[j] - cy, dz = pz[j] - cz;
      const float d = dx * dx + dy * dy + dz * dz;
      md[j] = fminf(md[j], d);
    }
  }
}

// ---------------------------------------------------------------------------
// Kernel 2: KNN via WMMA f32 16x16x4 + threshold-pruned top-32 selection.
// One workgroup (256 thr = 8 waves) per (batch, 16-center tile).
// score(m,n) = -2*c_m . x_n + |x_n|^2  (drops the per-center |c|^2 constant,
// which does not affect the top-k ordering).
//   A (16x4): row m = (-2cx, -2cy, -2cz, 1)  -> lanes<16 hold K0,K1; >=16 K2,K3
//   B (4x16): col n = (x, y, z, |x|^2)
// Score tiles go to an LDS buffer. Each wave owns 2 centers and keeps a
// register-resident sorted best-32 (lane r = rank r). Per chunk: one filter
// scan (score < current 32nd-best) compacts survivors into a per-wave LDS
// buffer via ds-atomics; each survivor is inserted with ballot+shfl_up.
// ---------------------------------------------------------------------------
__global__ __launch_bounds__(256) void knn_kernel(const float* __restrict__ xyz,
                                                  const float* __restrict__ center,
                                                  float* __restrict__ outNeigh,
                                                  float* __restrict__ outOrg) {
  __shared__ float sScores[16 * CHUNK];   // 16 KB
  __shared__ float sCandS[8 * CAP];       //  8 KB
  __shared__ int   sCandI[8 * CAP];       //  8 KB
  __shared__ int   sCnt[8];

  const int b    = blockIdx.x >> 5;          // batch
  const int g0   = (blockIdx.x & 31) << 4;   // first center of this tile
  const int t    = threadIdx.x;
  const int lane = t & 31;
  const int wave = t >> 5;
  const bool lo16 = (lane < 16);

  // Build the A fragment (same for every tile this workgroup computes).
  const float* cp0 = center + ((size_t)(b * NGROUP + g0 + (lane & 15))) * 3;
  const float cax = cp0[0], cay = cp0[1], caz = cp0[2];
  v2f afrag;
  afrag.x = lo16 ? (-2.0f * cax) : (-2.0f * caz);
  afrag.y = lo16 ? (-2.0f * cay) : 1.0f;

  // Register-resident sorted best lists: lane r holds rank-r (score, index).
  float bs0 = INFINITY, bs1 = INFINITY;
  int   bi0 = 0,        bi1 = 0;

  const float* xb = xyz + (size_t)b * NPTS * 3;

  // Filter + insert survivors for one center; best list stays in registers.
  auto select_center = [&](int c, int base, float& bs, int& bi) {
    if (lane == 0) sCnt[wave] = 0;                  // same-wave LDS is in-order
    const float thr = __shfl(bs, 31, 32);           // current 32nd-best
    for (int j = lane; j < CHUNK; j += 32) {
      const float s = sScores[c * CHUNK + j];
      if (s < thr) {
        const int slot = atomicAdd(&sCnt[wave], 1); // slot < CAP by construction
        sCandS[wave * CAP + slot] = s;
        sCandI[wave * CAP + slot] = base + j;
      }
    }
    const int K = sCnt[wave];
    for (int q = 0; q < K; ++q) {
      const float s = sCandS[wave * CAP + q];       // wave-uniform broadcast
      const int   i = sCandI[wave * CAP + q];
      const float worst = __shfl(bs, 31, 32);
      if (s < worst) {                              // wave-uniform branch
        const unsigned mask = (unsigned)__ballot(bs <= s);
        const int pos = __popc(mask);               // insert after equals
        const float upS = __shfl_up(bs, 1, 32);
        const int   upI = __shfl_up(bi, 1, 32);
        if (lane == pos)      { bs = s;   bi = i;   }
        else if (lane > pos)  { bs = upS; bi = upI; }
      }
    }
  };

  for (int chunk = 0; chunk < NCHUNK; ++chunk) {
    const int base = chunk * CHUNK;

    // ---- score CHUNK points for all 16 centers via WMMA --------------------
    for (int tile = wave; tile < CHUNK / 16; tile += 8) {
      const int p = base + tile * 16 + (lane & 15);
      const float px = xb[p * 3 + 0], py = xb[p * 3 + 1], pz = xb[p * 3 + 2];
      if (chunk + 1 < NCHUNK)                       // warm L2/L0 for next chunk
        __builtin_prefetch(&xb[(p + CHUNK) * 3], 0, 0);
      const float nrm = px * px + py * py + pz * pz;
      v2f bfrag;
      bfrag.x = lo16 ? px : pz;                     // cndmask, no EXEC split
      bfrag.y = lo16 ? py : nrm;
      v8f acc = {};
      acc = __builtin_amdgcn_wmma_f32_16x16x4_f32(
          /*neg_a=*/false, afrag, /*neg_b=*/false, bfrag,
          /*c_mod=*/(short)0, acc, /*reuse_a=*/false, /*reuse_b=*/false);
      const int ncol  = tile * 16 + (lane & 15);    // chunk-local column
      const int mbase = lo16 ? 0 : 8;
#pragma unroll
      for (int v = 0; v < 8; ++v)
        sScores[(mbase + v) * CHUNK + ncol] = acc[v];
    }
    __syncthreads();

    // ---- per-wave selection: wave owns centers 2w and 2w+1 -----------------
    select_center(wave * 2 + 0, base, bs0, bi0);
    select_center(wave * 2 + 1, base, bs1, bi1);
    __syncthreads();
  }

  // ---- emit outputs straight from registers: lane = rank r -----------------
#pragma unroll
  for (int cc = 0; cc < 2; ++cc) {
    const int g   = g0 + wave * 2 + cc;
    const int idx = (cc == 0) ? bi0 : bi1;
    const float px = xb[idx * 3 + 0], py = xb[idx * 3 + 1], pz = xb[idx * 3 + 2];
    const float* cp = center + ((size_t)(b * NGROUP + g)) * 3;
    const float ccx = cp[0], ccy = cp[1], ccz = cp[2];
    const size_t o = ((size_t)((b * NGROUP + g) * KNN) + lane) * 3;
    outOrg[o + 0] = px;         outOrg[o + 1] = py;         outOrg[o + 2] = pz;
    outNeigh[o + 0] = px - ccx; outNeigh[o + 1] = py - ccy; outNeigh[o + 2] = pz - ccz;
  }
}

// ---------------------------------------------------------------------------
extern "C" void kernel_launch(void* const* d_in, const int* in_sizes, int n_in,
                              void* d_out, int out_size, void* d_ws, size_t ws_size,
                              hipStream_t stream) {
  (void)in_sizes; (void)n_in; (void)out_size; (void)d_ws; (void)ws_size;
  const float* xyz = (const float*)d_in[0];
  float* out       = (float*)d_out;
  float* outNeigh  = out;                                  // [B,G,M,3]
  float* outCenter = out + NEIGH_ELEMS;                    // [B,G,3]
  float* outOrg    = out + NEIGH_ELEMS + CENTER_ELEMS;     // [B,G,M,3]

  fps_kernel<<<BATCH, 1024, 0, stream>>>(xyz, outCenter);
  knn_kernel<<<BATCH * (NGROUP / 16), 256, 0, stream>>>(xyz, outCenter,
                                                        outNeigh, outOrg);
}